// HetMSAMRGNN_17248588660764
// MI455X (gfx1250) — compile-verified
//
#include <hip/hip_runtime.h>
#include <hip/hip_bf16.h>

typedef __attribute__((ext_vector_type(2))) float v2f;
typedef __attribute__((ext_vector_type(8))) float v8f;

#define V_ 2
#define L_ 2
#define NP_ 120000
#define NC_ 120000
#define D_ 128
#define S_ 100
#define NNEG_ 50
#define INV_TEMP 2.0f            // 1 / 0.5
#define ROWS_PER_BLOCK 5400      // 100 anchors + 300 positives + 5000 cross-negs
#define PAD_ROWS 5408            // 338 tiles * 16
#define TILES_PER_BLOCK 338
#define NBLOCKS 8                // {p,c} x V x L
#define LDSSTRIDE 130            // 128 + 2 pad: rotates banks, keeps float2 align

// ---------------------------------------------------------------------------
// Kernel 1: gather the needed embedding rows, run the 2-layer MLP projector
// with fp32 WMMA (v_wmma_f32_16x16x4_f32), l2-normalize, write to workspace.
// One workgroup (8 waves) per 16-row tile; wave w owns output cols [16w,16w+16).
// ---------------------------------------------------------------------------
__global__ __launch_bounds__(256) void gather_project_kernel(
    const float* __restrict__ embP, const float* __restrict__ embC,
    const float* __restrict__ W1, const float* __restrict__ b1,
    const float* __restrict__ W2, const float* __restrict__ b2,
    const long long* __restrict__ idxP, const long long* __restrict__ idxC,
    const long long* __restrict__ negP, const long long* __restrict__ negC,
    float* __restrict__ proj)
{
    __shared__ float Xs[16][LDSSTRIDE];
    __shared__ float Hs[16][LDSSTRIDE];
    __shared__ float Os[16][LDSSTRIDE];
    __shared__ float rscale[16];
    __shared__ const float* srcPtr[16];

    const int tid  = threadIdx.x;
    const int blk  = blockIdx.x / TILES_PER_BLOCK;   // 0..7
    const int tile = blockIdx.x % TILES_PER_BLOCK;
    const int t    = blk >> 2;                       // 0 = p-anchors, 1 = c-anchors
    const int vi   = (blk >> 1) & 1;
    const int li   = blk & 1;
    const int row0 = tile * 16;

    const float* selfEmb      = t ? embC : embP;
    const float* otherEmb     = t ? embP : embC;
    const long long* selfIdx  = t ? idxC : idxP;
    const long long* negIdx   = t ? negC : negP;

    // Row -> source-row mapping for this tile (threads 0..15).
    if (tid < 16) {
        const int rg = row0 + tid;
        const float* p = nullptr;
        if (rg < ROWS_PER_BLOCK) {
            const int plane = vi * L_ + li;
            if (rg < 400) {
                const int k = rg / 100;              // 0 anchor, 1..3 positive keys
                const int s = rg - k * 100;
                int pv = vi, pl = li;
                if (k == 1)      { pv = 1 - vi; }
                else if (k == 2) { pl = 1 - li; }
                else if (k == 3) { pv = 1 - vi; pl = 1 - li; }
                const long long id = selfIdx[plane * S_ + s];
                p = selfEmb + ((size_t)(pv * L_ + pl) * NP_ + (size_t)id) * D_;
            } else {
                const int j = rg - 400;              // j = s*NNEG + n
                const long long id = negIdx[(size_t)plane * (S_ * NNEG_) + j];
                p = otherEmb + ((size_t)plane * NC_ + (size_t)id) * D_;
            }
        }
        srcPtr[tid] = p;
    }
    __syncthreads();

    // Gather 16 x 128 fp32 into LDS.
    #pragma unroll
    for (int i = 0; i < 8; ++i) {
        const int e = tid + i * 256;
        const int r = e >> 7;
        const int c = e & 127;
        const float* p = srcPtr[r];
        Xs[r][c] = p ? p[c] : 0.0f;
    }
    __syncthreads();

    const int wave = tid >> 5;
    const int lane = tid & 31;
    const int half = lane >> 4;       // 0: lanes 0-15, 1: lanes 16-31
    const int l16  = lane & 15;
    const int ncol = wave * 16 + l16; // output column owned by this lane

    // ---- H = relu(X @ W1 + b1), K = 128 in steps of 4 ----
    v8f acc = {};
    #pragma unroll
    for (int k0 = 0; k0 < D_; k0 += 4) {
        v2f a, b;
        // A 16x4 f32: lanes 0-15 hold K=k0,k0+1; lanes 16-31 hold K=k0+2,k0+3
        a.x = Xs[l16][k0 + 2 * half];
        a.y = Xs[l16][k0 + 2 * half + 1];
        // B 4x16 f32: N across lanes, same K split per half-wave
        b.x = W1[(k0 + 2 * half) * D_ + ncol];
        b.y = W1[(k0 + 2 * half + 1) * D_ + ncol];
        acc = __builtin_amdgcn_wmma_f32_16x16x4_f32(
            false, a, false, b, (short)0, acc, false, false);
    }
    {
        const float bias = b1[ncol];
        #pragma unroll
        for (int v = 0; v < 8; ++v) {
            const int m = v + 8 * half;          // C/D layout: VGPR v -> row m
            Hs[m][ncol] = fmaxf(acc[v] + bias, 0.0f);
        }
    }
    __syncthreads();

    // ---- O = H @ W2 + b2 ----
    v8f acc2 = {};
    #pragma unroll
    for (int k0 = 0; k0 < D_; k0 += 4) {
        v2f a, b;
        a.x = Hs[l16][k0 + 2 * half];
        a.y = Hs[l16][k0 + 2 * half + 1];
        b.x = W2[(k0 + 2 * half) * D_ + ncol];
        b.y = W2[(k0 + 2 * half + 1) * D_ + ncol];
        acc2 = __builtin_amdgcn_wmma_f32_16x16x4_f32(
            false, a, false, b, (short)0, acc2, false, false);
    }
    {
        const float bias = b2[ncol];
        #pragma unroll
        for (int v = 0; v < 8; ++v) {
            const int m = v + 8 * half;
            Os[m][ncol] = acc2[v] + bias;
        }
    }
    __syncthreads();

    // Row L2 norms (x / max(||x||, 1e-12)).
    if (tid < 16) {
        float ss = 0.0f;
        for (int c = 0; c < D_; ++c) { const float x = Os[tid][c]; ss += x * x; }
        rscale[tid] = 1.0f / fmaxf(sqrtf(ss), 1e-12f);
    }
    __syncthreads();

    #pragma unroll
    for (int i = 0; i < 8; ++i) {
        const int e = tid + i * 256;
        const int r = e >> 7;
        const int c = e & 127;
        const int rg = row0 + r;
        if (rg < ROWS_PER_BLOCK)
            proj[((size_t)blk * PAD_ROWS + rg) * D_ + c] = Os[r][c] * rscale[r];
    }
}

// ---------------------------------------------------------------------------
// Kernel 2: one wave per (block, anchor). 152 dot products of length 128:
// float4 per lane + shfl_xor butterfly, exp(2*dot) sums, -log(pos/(pos+neg)).
// ---------------------------------------------------------------------------
__global__ __launch_bounds__(32) void loss_kernel(
    const float* __restrict__ proj, float* __restrict__ contrib)
{
    const int blk = blockIdx.x / S_;
    const int s   = blockIdx.x % S_;
    const float* base = proj + (size_t)blk * PAD_ROWS * D_;
    const float* Z   = base;             // [100][128] anchors
    const float* POS = base + 100 * D_;  // [3][100][128]
    const float* NEG = base + 400 * D_;  // [100*50][128]
    const int lane = threadIdx.x;
    const int c0   = lane * 4;

    const float4 z = *(const float4*)(Z + (size_t)s * D_ + c0);

    auto dot = [&](const float* v) -> float {
        const float4 w = *(const float4*)(v + c0);
        float p = z.x * w.x + z.y * w.y + z.z * w.z + z.w * w.w;
        #pragma unroll
        for (int off = 16; off > 0; off >>= 1) p += __shfl_xor(p, off, 32);
        return p;
    };

    float possum = 0.0f;
    #pragma unroll
    for (int k = 0; k < 3; ++k)
        possum += expf(INV_TEMP * dot(POS + (size_t)(k * S_ + s) * D_));

    float negsum = 0.0f;
    for (int j = 0; j < S_; ++j) {       // within-type, diagonal skipped
        if (j == s) continue;
        negsum += expf(INV_TEMP * dot(Z + (size_t)j * D_));
    }
    for (int n = 0; n < NNEG_; ++n)      // cross-type sampled negatives
        negsum += expf(INV_TEMP * dot(NEG + (size_t)(s * NNEG_ + n) * D_));

    if (lane == 0)
        contrib[blockIdx.x] = -logf(possum / (possum + negsum));
}

// ---------------------------------------------------------------------------
// Kernel 3: deterministic fixed-order reduction of 800 contributions.
// ---------------------------------------------------------------------------
__global__ __launch_bounds__(256) void reduce_kernel(
    const float* __restrict__ contrib, float* __restrict__ out)
{
    __shared__ float sdata[256];
    const int tid = threadIdx.x;
    float v = 0.0f;
    for (int i = tid; i < NBLOCKS * S_; i += 256) v += contrib[i];
    sdata[tid] = v;
    __syncthreads();
    for (int step = 128; step > 0; step >>= 1) {
        if (tid < step) sdata[tid] += sdata[tid + step];
        __syncthreads();
    }
    if (tid == 0) out[0] = sdata[0] / (float)(NBLOCKS * S_);
}

extern "C" void kernel_launch(void* const* d_in, const int* in_sizes, int n_in,
                              void* d_out, int out_size, void* d_ws, size_t ws_size,
                              hipStream_t stream) {
    (void)in_sizes; (void)n_in; (void)out_size; (void)ws_size;
    const float* embP = (const float*)d_in[0];
    const float* embC = (const float*)d_in[1];
    const float* W1   = (const float*)d_in[2];
    const float* b1   = (const float*)d_in[3];
    const float* W2   = (const float*)d_in[4];
    const float* b2   = (const float*)d_in[5];
    const long long* idxP = (const long long*)d_in[6];
    const long long* idxC = (const long long*)d_in[7];
    const long long* negP = (const long long*)d_in[8];
    const long long* negC = (const long long*)d_in[9];

    float* proj    = (float*)d_ws;                               // 8*5408*128 f32
    float* contrib = proj + (size_t)NBLOCKS * PAD_ROWS * D_;     // 800 f32

    gather_project_kernel<<<NBLOCKS * TILES_PER_BLOCK, 256, 0, stream>>>(
        embP, embC, W1, b1, W2, b2, idxP, idxC, negP, negC, proj);
    loss_kernel<<<NBLOCKS * S_, 32, 0, stream>>>(proj, contrib);
    reduce_kernel<<<1, 256, 0, stream>>>(contrib, (float*)d_out);
}